// GINE_50036368998499
// MI455X (gfx1250) — compile-verified
//
#include <hip/hip_runtime.h>
#include <hip/hip_bf16.h>

typedef float v2f __attribute__((ext_vector_type(2)));
typedef float v8f __attribute__((ext_vector_type(8)));

#define H 64
#define NGRAPHS 64
#define WSTRIDE 160   // packed-pair row stride in dwords: 128 + 32 pad (bank-conflict-free)

// ---------------- embedding: x = relu(pos @ emb_w + emb_b) ----------------
__global__ void k_embed(const float* __restrict__ pos,
                        const float* __restrict__ w, const float* __restrict__ b,
                        float* __restrict__ x, int N) {
  int t = blockIdx.x * blockDim.x + threadIdx.x;
  if (t >= N * H) return;
  int i = t >> 6, c = t & 63;
  float p0 = pos[2 * i], p1 = pos[2 * i + 1];
  float v = fmaf(p0, w[c], fmaf(p1, w[H + c], b[c]));
  x[t] = fmaxf(v, 0.f);
}

// ---- edge phase: agg[dst] += relu(h[src] + edge_attr*le_w + le_b) --------
// agg is pre-initialized with h, so after this kernel agg = h + sum(messages).
// 16 threads per edge, 4 features (float4) per thread.
__global__ void k_edge(const float* __restrict__ h,
                       const int* __restrict__ ei, const float* __restrict__ ea,
                       const float* __restrict__ lw, const float* __restrict__ lb,
                       float* __restrict__ agg, int E) {
  long long t = (long long)blockIdx.x * blockDim.x + threadIdx.x;
  int e = (int)(t >> 4);
  if (e >= E) return;
  int c4 = ((int)t & 15) << 2;
  int src = ei[e];
  int dst = ei[E + e];
  float a = ea[e];
  float4 w  = *(const float4*)(lw + c4);
  float4 bb = *(const float4*)(lb + c4);
  float4 xv = *(const float4*)(h + (long long)src * H + c4);
  float m0 = fmaxf(xv.x + fmaf(a, w.x, bb.x), 0.f);
  float m1 = fmaxf(xv.y + fmaf(a, w.y, bb.y), 0.f);
  float m2 = fmaxf(xv.z + fmaf(a, w.z, bb.z), 0.f);
  float m3 = fmaxf(xv.w + fmaf(a, w.w, bb.w), 0.f);
  float* ap = agg + (long long)dst * H + c4;
  atomicAdd(ap + 0, m0);
  atomicAdd(ap + 1, m1);
  atomicAdd(ap + 2, m2);
  atomicAdd(ap + 3, m3);
}

// ---- node GEMM: y = relu(z @ W + bias), fused PairNorm stats -------------
// wave32 WMMA f32 16x16x4: each wave computes a 16-row x 64-col tile
// (4 v8f accumulators), K=64 in 16 steps.
// W is async-scattered into a packed-pair LDS layout
//   wl[(k>>1)*WSTRIDE + 2n + (k&1)] = W[k][n]
// via per-lane GLOBAL_LOAD_ASYNC_TO_LDS_B32 (ASYNCcnt), so each B fragment
// is one 8-byte ds_load_b64 straight into an even VGPR pair (no repacking
// movs), and the +32-dword row pad makes the b64 reads bank-conflict-free.
__global__ __launch_bounds__(256)
void k_gemm(const float* __restrict__ z,
            const float* __restrict__ W, const float* __restrict__ bias,
            float* __restrict__ y, float* __restrict__ stats,
            int Nrows, int do_relu) {
  __shared__ __attribute__((aligned(16))) float wl[32 * WSTRIDE];
  __shared__ __attribute__((aligned(16))) float bl[H];

  // async stage: 4096 floats of W packed + 64 floats of bias into LDS.
  {
    int t = threadIdx.x;
    int k0t = t >> 6;            // 0..3 : k sub-index for this thread
    int n0  = t & 63;
    // packed LDS dword offset for element (k = k0t, n = n0)
    unsigned pbase = (unsigned)((k0t >> 1) * WSTRIDE + 2 * n0 + (k0t & 1));
    // generic LDS pointer -> low 32 bits = LDS byte offset (flat aperture rule)
    unsigned lds0 = (unsigned)(uintptr_t)(void*)wl + pbase * 4u;
    unsigned long long g0 = (unsigned long long)(uintptr_t)W + (unsigned long long)t * 4ull;
    #pragma unroll
    for (int r = 0; r < 16; ++r) {
      // element index idx = r*256 + t : k += 4 per step -> LDS += 2*WSTRIDE
      unsigned lds = lds0 + (unsigned)r * (2u * WSTRIDE * 4u);   // +1280 B
      unsigned long long ga = g0 + (unsigned long long)r * 1024ull;
      asm volatile("global_load_async_to_lds_b32 %0, %1, off"
                   :: "v"(lds), "v"(ga) : "memory");
    }
    if (t < 16) {
      unsigned bl_lds = (unsigned)(uintptr_t)(void*)bl + (unsigned)t * 16u;
      unsigned long long gb = (unsigned long long)(uintptr_t)bias + (unsigned long long)t * 16ull;
      asm volatile("global_load_async_to_lds_b128 %0, %1, off"
                   :: "v"(bl_lds), "v"(gb) : "memory");
    }
    asm volatile("s_wait_asynccnt 0x0" ::: "memory");
  }
  __syncthreads();

  int lane = threadIdx.x & 31;
  int wave = threadIdx.x >> 5;
  int rowTile = blockIdx.x * 8 + wave;
  if (rowTile * 16 >= Nrows) return;

  int m = lane & 15;              // A row within tile
  int n = lane & 15;              // B/D col within 16-wide subtile
  int kq = (lane >> 4) << 1;      // K sub-offset: lanes 16-31 handle K+2,K+3
  long long rb = (long long)(rowTile * 16 + m) * H;

  v8f acc0 = {}, acc1 = {}, acc2 = {}, acc3 = {};
  #pragma unroll
  for (int k0 = 0; k0 < H; k0 += 4) {
    int ka = k0 + kq;             // even
    v2f a = *(const v2f*)(z + rb + ka);                 // global b64, 8B aligned
    const float* wp = wl + ((ka >> 1) * WSTRIDE + 2 * n);
    v2f b0 = *(const v2f*)(wp);                         // ds_load_b64
    v2f b1 = *(const v2f*)(wp + 32);
    v2f b2 = *(const v2f*)(wp + 64);
    v2f b3 = *(const v2f*)(wp + 96);
    acc0 = __builtin_amdgcn_wmma_f32_16x16x4_f32(false, a, false, b0, (short)0, acc0, false, false);
    acc1 = __builtin_amdgcn_wmma_f32_16x16x4_f32(false, a, false, b1, (short)0, acc1, false, false);
    acc2 = __builtin_amdgcn_wmma_f32_16x16x4_f32(false, a, false, b2, (short)0, acc2, false, false);
    acc3 = __builtin_amdgcn_wmma_f32_16x16x4_f32(false, a, false, b3, (short)0, acc3, false, false);
  }

  // D layout: vgpr j -> M = j (lanes 0-15) or j+8 (lanes 16-31); N = lane%16
  int mbase = rowTile * 16 + ((lane >> 4) << 3);
  float bn0 = bl[n], bn1 = bl[n + 16], bn2 = bl[n + 32], bn3 = bl[n + 48];
  float s0 = 0.f, s1 = 0.f, s2 = 0.f, s3 = 0.f, q = 0.f;
  #pragma unroll
  for (int j = 0; j < 8; ++j) {
    long long orow = (long long)(mbase + j) * H;
    float v0 = acc0[j] + bn0;
    float v1 = acc1[j] + bn1;
    float v2 = acc2[j] + bn2;
    float v3 = acc3[j] + bn3;
    if (do_relu) {
      v0 = fmaxf(v0, 0.f); v1 = fmaxf(v1, 0.f);
      v2 = fmaxf(v2, 0.f); v3 = fmaxf(v3, 0.f);
    }
    y[orow + n]      = v0;
    y[orow + n + 16] = v1;
    y[orow + n + 32] = v2;
    y[orow + n + 48] = v3;
    s0 += v0; s1 += v1; s2 += v2; s3 += v3;
    q = fmaf(v0, v0, q); q = fmaf(v1, v1, q);
    q = fmaf(v2, v2, q); q = fmaf(v3, v3, q);
  }
  if (stats) {
    atomicAdd(stats + n,      s0);
    atomicAdd(stats + n + 16, s1);
    atomicAdd(stats + n + 32, s2);
    atomicAdd(stats + n + 48, s3);
    atomicAdd(stats + H,      q);  // total sum of squares
  }
}

// ---- PairNorm finalize: mu[h] = S[h]/N ; inv_scale from Q - sum(S^2)/N ---
__global__ void k_finalize(const float* __restrict__ stats,
                           float* __restrict__ ms, float invN) {
  __shared__ float red[H];
  int t = threadIdx.x;  // 64 threads
  float s = stats[t];
  ms[t] = s * invN;     // mu
  red[t] = s * s;
  __syncthreads();
  for (int o = 32; o > 0; o >>= 1) {
    if (t < o) red[t] += red[t + o];
    __syncthreads();
  }
  if (t == 0) {
    float Q = stats[H];
    float var = (Q - red[0] * invN) * invN;
    ms[H] = rsqrtf(1e-5f + var);
  }
}

__global__ void k_apply(float* __restrict__ y, const float* __restrict__ ms, int N) {
  int t = blockIdx.x * blockDim.x + threadIdx.x;
  if (t >= N * H) return;
  int c = t & 63;
  y[t] = (y[t] - ms[c]) * ms[H];
}

// ---------------- global max pool -----------------------------------------
__global__ void k_initpool(float* __restrict__ g) {
  int t = blockIdx.x * blockDim.x + threadIdx.x;
  if (t < NGRAPHS * H) g[t] = -__builtin_inff();
}

__device__ __forceinline__ void atomicMaxF(float* addr, float v) {
  // works for any sign given -inf initialization
  if (v >= 0.f) atomicMax((int*)addr, __float_as_int(v));
  else          atomicMin((unsigned int*)addr, (unsigned int)__float_as_int(v));
}

__global__ void k_pool(const float* __restrict__ h, const int* __restrict__ batch,
                       float* __restrict__ g, int N) {
  int t = blockIdx.x * blockDim.x + threadIdx.x;
  if (t >= N * H) return;
  int i = t >> 6, c = t & 63;
  int b = batch[i];
  atomicMaxF(g + b * H + c, h[t]);
}

// ---------------- final [64,64] x [64,2] linear ---------------------------
__global__ void k_mlp2(const float* __restrict__ g1, const float* __restrict__ w,
                       const float* __restrict__ b, float* __restrict__ out) {
  int t = threadIdx.x;  // 128 threads
  int r = t >> 1, c = t & 1;
  float acc = b[c];
  for (int k = 0; k < H; ++k) acc = fmaf(g1[r * H + k], w[k * 2 + c], acc);
  out[t] = acc;
}

extern "C" void kernel_launch(void* const* d_in, const int* in_sizes, int n_in,
                              void* d_out, int out_size, void* d_ws, size_t ws_size,
                              hipStream_t stream) {
  (void)n_in; (void)out_size; (void)ws_size;
  const float* pos    = (const float*)d_in[0];
  const int*   ei     = (const int*)d_in[1];
  const float* ea     = (const float*)d_in[2];
  const int*   batch  = (const int*)d_in[3];
  const float* emb_w  = (const float*)d_in[4];
  const float* emb_b  = (const float*)d_in[5];
  const float* le_w[3] = {(const float*)d_in[6],  (const float*)d_in[10], (const float*)d_in[14]};
  const float* le_b[3] = {(const float*)d_in[7],  (const float*)d_in[11], (const float*)d_in[15]};
  const float* nn_w[3] = {(const float*)d_in[8],  (const float*)d_in[12], (const float*)d_in[16]};
  const float* nn_b[3] = {(const float*)d_in[9],  (const float*)d_in[13], (const float*)d_in[17]};
  const float* lin1_w = (const float*)d_in[18];
  const float* lin1_b = (const float*)d_in[19];
  const float* lin2_w = (const float*)d_in[20];
  const float* lin2_b = (const float*)d_in[21];

  const int N = in_sizes[0] / 2;   // 50000
  const int E = in_sizes[2];       // 800000

  // workspace layout (floats)
  float* ws    = (float*)d_ws;
  size_t NH    = (size_t)N * H;
  float* buf0  = ws;               // features (ping)
  float* agg   = buf0 + NH;        // z = x + sum(messages)
  float* buf2  = agg + NH;         // features (pong)
  float* stats = buf2 + NH;        // S[64] + Q  (padded to 80)
  float* ms    = stats + 80;       // mu[64] + inv_scale (padded to 80)
  float* g     = ms + 80;          // pooled [64,64]
  float* g1    = g + NGRAPHS * H;  // mlp hidden [64,64]

  const int threads = 256;
  const int nh_blocks = (int)((NH + threads - 1) / threads);
  const int rowTiles = (N + 15) / 16;
  const int gemm_blocks = (rowTiles + 7) / 8;
  const long long et = (long long)E * 16;
  const int edge_blocks = (int)((et + threads - 1) / threads);

  k_embed<<<nh_blocks, threads, 0, stream>>>(pos, emb_w, emb_b, buf0, N);

  float* hin = buf0;
  float* hout = buf2;
  for (int l = 0; l < 3; ++l) {
    // agg := hin, so edge atomics accumulate on top of node features
    hipMemcpyAsync(agg, hin, NH * sizeof(float), hipMemcpyDeviceToDevice, stream);
    hipMemsetAsync(stats, 0, 80 * sizeof(float), stream);
    k_edge<<<edge_blocks, threads, 0, stream>>>(hin, ei, ea, le_w[l], le_b[l], agg, E);
    k_gemm<<<gemm_blocks, threads, 0, stream>>>(agg, nn_w[l], nn_b[l], hout, stats, N, 1);
    k_finalize<<<1, 64, 0, stream>>>(stats, ms, 1.0f / (float)N);
    k_apply<<<nh_blocks, threads, 0, stream>>>(hout, ms, N);
    float* tmp = hin; hin = hout; hout = tmp;
  }

  k_initpool<<<(NGRAPHS * H + threads - 1) / threads, threads, 0, stream>>>(g);
  k_pool<<<nh_blocks, threads, 0, stream>>>(hin, batch, g, N);
  // g1 = relu(g @ lin1_w + lin1_b)  — reuse WMMA GEMM (4 row tiles, 1 block)
  k_gemm<<<1, threads, 0, stream>>>(g, lin1_w, lin1_b, g1, nullptr, NGRAPHS, 1);
  k_mlp2<<<1, 128, 0, stream>>>(g1, lin2_w, lin2_b, (float*)d_out);
}